// MyModel_25769804311
// MI455X (gfx1250) — compile-verified
//
#include <hip/hip_runtime.h>
#include <math.h>

typedef float v2f __attribute__((ext_vector_type(2)));
typedef float v8f __attribute__((ext_vector_type(8)));

// D = A(16x4 f32) * B(4x16 f32) + C(16x16 f32)
#define WMMA_F32X4(a, b, c) \
  __builtin_amdgcn_wmma_f32_16x16x4_f32(false, (a), false, (b), (short)0, (c), false, false)

// Shared epilogue: write relu(c[v] + bias[col]) for the 16x16 tile.
// `full` is wave-uniform -> single branch, straight-line stores on the hot path.
__device__ __forceinline__ void store_tile_relu(v8f c, const float* __restrict__ bias,
                                                float* __restrict__ out, int tile, int n,
                                                int lane) {
  int col = lane & 15;
  int rbase = tile * 16 + ((lane < 16) ? 0 : 8);
  float bv = bias[col];
  bool full = (tile * 16 + 16) <= n;  // uniform across the wave
  if (full) {
#pragma unroll
    for (int v = 0; v < 8; ++v) {
      float s = c[v] + bv;
      out[(long long)(rbase + v) * 16 + col] = s > 0.f ? s : 0.f;
    }
  } else {
#pragma unroll
    for (int v = 0; v < 8; ++v) {
      int r = rbase + v;
      if (r < n) {
        float s = c[v] + bv;
        out[(long long)r * 16 + col] = s > 0.f ? s : 0.f;
      }
    }
  }
}

// ---------------- Layer-1 projection: h = relu(x @ p1w + p1b), din=3, padded to stride 4
__global__ void proj1_kernel(const float* __restrict__ x, const float* __restrict__ pw,
                             const float* __restrict__ pb, float* __restrict__ h, int n) {
  int i = blockIdx.x * blockDim.x + threadIdx.x;
  if (i >= n) return;
  float x0 = x[(long long)i * 3 + 0];
  float x1 = x[(long long)i * 3 + 1];
  float x2 = x[(long long)i * 3 + 2];
#pragma unroll
  for (int c = 0; c < 3; ++c) {
    float s = fmaf(x0, pw[0 * 3 + c], fmaf(x1, pw[1 * 3 + c], fmaf(x2, pw[2 * 3 + c], pb[c])));
    h[(long long)i * 4 + c] = s > 0.f ? s : 0.f;
  }
  h[(long long)i * 4 + 3] = 0.f;
}

// ---------------- Edge scatter, stride-4 features (layer 1, 3 real features)
__global__ void edge_scatter4(const int* __restrict__ eidx, long long E,
                              const float* __restrict__ h, float* __restrict__ agg) {
  long long t = (long long)blockIdx.x * blockDim.x + threadIdx.x;
  long long e = t >> 2;
  int c = (int)(t & 3);
  if (e >= E || c >= 3) return;
  int src = eidx[2 * e];            // low dword of int64 (values < 2^31)
  int dst = eidx[2 * (E + e)];
  atomicAdd(&agg[(long long)dst * 4 + c], h[(long long)src * 4 + c]);
}

// ---------------- Edge scatter, stride-16 features (layers 2,3)
__global__ void edge_scatter16(const int* __restrict__ eidx, long long E,
                               const float* __restrict__ h, float* __restrict__ agg) {
  long long t = (long long)blockIdx.x * blockDim.x + threadIdx.x;
  long long e = t >> 4;
  int c = (int)(t & 15);
  if (e >= E) return;
  int src = eidx[2 * e];
  int dst = eidx[2 * (E + e)];
  atomicAdd(&agg[(long long)dst * 16 + c], h[(long long)src * 16 + c]);
}

// ---------------- WMMA GEMM: out = relu(xin[N,16] @ w[16,16] + bias), one wave per 16-row tile
__global__ void gemm16_wmma(const float* __restrict__ xin, const float* __restrict__ w,
                            const float* __restrict__ bias, float* __restrict__ out, int n) {
  int lane = threadIdx.x & 31;
  int wave = threadIdx.x >> 5;
  int tile = blockIdx.x * (blockDim.x >> 5) + wave;
  int nt = (n + 15) >> 4;
  if (tile >= nt) return;                 // wave-uniform: EXEC stays all-ones for live waves
  int row = tile * 16 + (lane & 15);      // A-matrix: M = lane%16 for both half-waves
  int rc = row < n ? row : n - 1;
  int col = lane & 15;                    // B/C/D: N = lane%16
  int kq = (lane < 16) ? 0 : 2;           // lanes 0-15 hold K=0,1; lanes 16-31 hold K=2,3
  v8f c = {0.f, 0.f, 0.f, 0.f, 0.f, 0.f, 0.f, 0.f};
#pragma unroll
  for (int kb = 0; kb < 16; kb += 4) {
    v2f a, b;
    a.x = xin[(long long)rc * 16 + kb + kq];
    a.y = xin[(long long)rc * 16 + kb + kq + 1];
    b.x = w[(kb + kq) * 16 + col];
    b.y = w[(kb + kq + 1) * 16 + col];
    c = WMMA_F32X4(a, b, c);
  }
  store_tile_relu(c, bias, out, tile, n, lane);
}

// ---------------- Layer-1 combine: out = relu(agg4[N,4] @ l1w(3x16) + l1b + x[N,3] @ r1w(3x16))
__global__ void combine1_wmma(const float* __restrict__ agg4, const float* __restrict__ lw,
                              const float* __restrict__ lb, const float* __restrict__ x,
                              const float* __restrict__ rw, float* __restrict__ out, int n) {
  int lane = threadIdx.x & 31;
  int wave = threadIdx.x >> 5;
  int tile = blockIdx.x * (blockDim.x >> 5) + wave;
  int nt = (n + 15) >> 4;
  if (tile >= nt) return;
  int row = tile * 16 + (lane & 15);
  int rc = row < n ? row : n - 1;
  int col = lane & 15;
  int kq = (lane < 16) ? 0 : 2;
  v8f c = {0.f, 0.f, 0.f, 0.f, 0.f, 0.f, 0.f, 0.f};
  v2f a, b;
  // agg @ lw   (K=3 padded to 4; agg[...,3] is zero from memset)
  a.x = agg4[(long long)rc * 4 + kq];
  a.y = agg4[(long long)rc * 4 + kq + 1];
  b.x = (kq < 3) ? lw[kq * 16 + col] : 0.f;
  b.y = (kq + 1 < 3) ? lw[(kq + 1) * 16 + col] : 0.f;
  c = WMMA_F32X4(a, b, c);
  // x @ rw
  a.x = (kq < 3) ? x[(long long)rc * 3 + kq] : 0.f;
  a.y = (kq + 1 < 3) ? x[(long long)rc * 3 + kq + 1] : 0.f;
  b.x = (kq < 3) ? rw[kq * 16 + col] : 0.f;
  b.y = (kq + 1 < 3) ? rw[(kq + 1) * 16 + col] : 0.f;
  c = WMMA_F32X4(a, b, c);
  store_tile_relu(c, lb, out, tile, n, lane);
}

// ---------------- Layer-2 combine: out = relu(agg[N,16]@lw + lb + xin[N,16]@rw)
__global__ void combine16_wmma(const float* __restrict__ agg, const float* __restrict__ lw,
                               const float* __restrict__ lb, const float* __restrict__ xin,
                               const float* __restrict__ rw, float* __restrict__ out, int n) {
  int lane = threadIdx.x & 31;
  int wave = threadIdx.x >> 5;
  int tile = blockIdx.x * (blockDim.x >> 5) + wave;
  int nt = (n + 15) >> 4;
  if (tile >= nt) return;
  int row = tile * 16 + (lane & 15);
  int rc = row < n ? row : n - 1;
  int col = lane & 15;
  int kq = (lane < 16) ? 0 : 2;
  v8f c = {0.f, 0.f, 0.f, 0.f, 0.f, 0.f, 0.f, 0.f};
#pragma unroll
  for (int kb = 0; kb < 16; kb += 4) {
    v2f a, b;
    a.x = agg[(long long)rc * 16 + kb + kq];
    a.y = agg[(long long)rc * 16 + kb + kq + 1];
    b.x = lw[(kb + kq) * 16 + col];
    b.y = lw[(kb + kq + 1) * 16 + col];
    c = WMMA_F32X4(a, b, c);
  }
#pragma unroll
  for (int kb = 0; kb < 16; kb += 4) {
    v2f a, b;
    a.x = xin[(long long)rc * 16 + kb + kq];
    a.y = xin[(long long)rc * 16 + kb + kq + 1];
    b.x = rw[(kb + kq) * 16 + col];
    b.y = rw[(kb + kq + 1) * 16 + col];
    c = WMMA_F32X4(a, b, c);
  }
  store_tile_relu(c, lb, out, tile, n, lane);
}

// ---------------- Layer-3 combine: out[n] = sigmoid(agg[n,:]@lw(16,1) + lb + xin[n,:]@rw(16,1))
__global__ void combine3_kernel(const float* __restrict__ agg, const float* __restrict__ lw,
                                const float* __restrict__ lb, const float* __restrict__ xin,
                                const float* __restrict__ rw, float* __restrict__ out, int n) {
  int i = blockIdx.x * blockDim.x + threadIdx.x;
  if (i >= n) return;
  float s = lb[0];
#pragma unroll
  for (int c = 0; c < 16; ++c) {
    s = fmaf(agg[(long long)i * 16 + c], lw[c], s);
    s = fmaf(xin[(long long)i * 16 + c], rw[c], s);
  }
  out[i] = 1.0f / (1.0f + expf(-s));
}

extern "C" void kernel_launch(void* const* d_in, const int* in_sizes, int n_in,
                              void* d_out, int out_size, void* d_ws, size_t ws_size,
                              hipStream_t stream) {
  const float* x = (const float*)d_in[0];
  const int* eidx = (const int*)d_in[1];  // int64 edge_index viewed as int pairs (LE low word)
  const float* p1w = (const float*)d_in[2];
  const float* p1b = (const float*)d_in[3];
  const float* l1w = (const float*)d_in[4];
  const float* l1b = (const float*)d_in[5];
  const float* r1w = (const float*)d_in[6];
  const float* p2w = (const float*)d_in[7];
  const float* p2b = (const float*)d_in[8];
  const float* l2w = (const float*)d_in[9];
  const float* l2b = (const float*)d_in[10];
  const float* r2w = (const float*)d_in[11];
  const float* p3w = (const float*)d_in[12];
  const float* p3b = (const float*)d_in[13];
  const float* l3w = (const float*)d_in[14];
  const float* l3b = (const float*)d_in[15];
  const float* r3w = (const float*)d_in[16];

  long long N = in_sizes[0] / 3;
  long long E = in_sizes[1] / 2;

  float* buf0 = (float*)d_ws;          // N*16 floats
  float* buf1 = buf0 + N * 16;         // N*16 floats
  float* buf2 = buf1 + N * 16;         // N*16 floats  (total 96 MB)

  dim3 blk(256);
  int nodeBlocks = (int)((N + 255) / 256);
  int nTiles = (int)((N + 15) / 16);
  int tileBlocks = (nTiles + 7) / 8;   // 8 waves (tiles) per 256-thread block
  int edge4Blocks = (int)((E * 4 + 255) / 256);
  int edge16Blocks = (int)((E * 16 + 255) / 256);

  // ================= Layer 1 (din=3 -> dout=16) =================
  proj1_kernel<<<nodeBlocks, blk, 0, stream>>>(x, p1w, p1b, buf0, (int)N);
  hipMemsetAsync(buf1, 0, (size_t)N * 4 * sizeof(float), stream);
  edge_scatter4<<<edge4Blocks, blk, 0, stream>>>(eidx, E, buf0, buf1);
  combine1_wmma<<<tileBlocks, blk, 0, stream>>>(buf1, l1w, l1b, x, r1w, buf2, (int)N);

  // ================= Layer 2 (16 -> 16), input buf2 =================
  gemm16_wmma<<<tileBlocks, blk, 0, stream>>>(buf2, p2w, p2b, buf0, (int)N);   // h -> buf0
  hipMemsetAsync(buf1, 0, (size_t)N * 16 * sizeof(float), stream);
  edge_scatter16<<<edge16Blocks, blk, 0, stream>>>(eidx, E, buf0, buf1);
  combine16_wmma<<<tileBlocks, blk, 0, stream>>>(buf1, l2w, l2b, buf2, r2w, buf0, (int)N); // out -> buf0

  // ================= Layer 3 (16 -> 1), input buf0 =================
  gemm16_wmma<<<tileBlocks, blk, 0, stream>>>(buf0, p3w, p3b, buf2, (int)N);   // h -> buf2
  hipMemsetAsync(buf1, 0, (size_t)N * 16 * sizeof(float), stream);
  edge_scatter16<<<edge16Blocks, blk, 0, stream>>>(eidx, E, buf2, buf1);
  combine3_kernel<<<nodeBlocks, blk, 0, stream>>>(buf1, l3w, l3b, buf0, r3w, (float*)d_out, (int)N);
}